// SGCN_76484777607282
// MI455X (gfx1250) — compile-verified
//
#include <hip/hip_runtime.h>

typedef __attribute__((ext_vector_type(2))) float v2f;
typedef __attribute__((ext_vector_type(8))) float v8f;

#define F 64

// ---------------- zero fill ----------------
__global__ void k_zero(float* __restrict__ p, long n) {
  long i = (long)blockIdx.x * blockDim.x + threadIdx.x;
  long stride = (long)gridDim.x * blockDim.x;
  for (; i < n; i += stride) p[i] = 0.0f;
}

// ---------------- weighted + unweighted in-degree ----------------
__global__ void k_edge_deg(const float* __restrict__ ew, const int* __restrict__ dst,
                           float* __restrict__ deg_w, float* __restrict__ deg, int E) {
  int e = blockIdx.x * blockDim.x + threadIdx.x;
  if (e >= E) return;
  int d = dst[e];
  atomicAdd(&deg_w[d], ew[e]);
  atomicAdd(&deg[d], 1.0f);
}

// ---------------- per-graph node counts ----------------
__global__ void k_counts(const int* __restrict__ gids, float* __restrict__ counts, int N) {
  int i = blockIdx.x * blockDim.x + threadIdx.x;
  if (i >= N) return;
  atomicAdd(&counts[gids[i]], 1.0f);
}

// ---------------- SpMM: neigh[dst] += (ew/deg_w[dst]) * h[src] ----------------
// one wave32 per edge; lane l handles features [2l, 2l+1] (float2 -> b64 loads)
__global__ void k_spmm(const float* __restrict__ h, const float* __restrict__ ew,
                       const float* __restrict__ deg_w,
                       const int* __restrict__ src, const int* __restrict__ dst,
                       float* __restrict__ neigh, int E) {
  int e = blockIdx.x * 8 + (threadIdx.x >> 5);
  if (e >= E) return;
  int lane = threadIdx.x & 31;
  int s = src[e], d = dst[e];
  float nw = ew[e] / deg_w[d];
  const float2* hs = reinterpret_cast<const float2*>(h + (long)s * F);
  float2 v = hs[lane];
  float* nb = neigh + (long)d * F + lane * 2;
  atomicAdd(nb,     v.x * nw);
  atomicAdd(nb + 1, v.y * nw);
}

// ---------------- fused node update + [128x64]x[64x64] WMMA GEMM ----------------
// hn = (neigh + h) / (deg + 1);  out = relu(hn @ W + b)
// readout==0: store rows to `out`; readout==1: atomicAdd into hg_sum[graph_id]
//
// LDS tiles are stored K-pair-interleaved as float2 so that every WMMA A/B
// fragment is one aligned ds_load_b64 feeding v_wmma directly (no repacking).
__global__ void __launch_bounds__(256)
k_update_gemm(const float* __restrict__ h, const float* __restrict__ neigh,
              const float* __restrict__ deg, const float* __restrict__ Wm,
              const float* __restrict__ bias, float* __restrict__ out,
              float* __restrict__ hg_sum, const int* __restrict__ gids,
              int n_nodes, int readout) {
  __shared__ v2f sA2[128][33];  // sA2[r][k2] = {hn[r][2k2], hn[r][2k2+1]}, k2 in [0,32)
  __shared__ v2f sW2[32][65];   // sW2[k2][c] = {W[2k2][c], W[2k2+1][c]},  c  in [0,64)

  const int tid  = threadIdx.x;
  const int base = blockIdx.x * 128;

  // stage W (64x64 row-major) into pair-interleaved layout
  for (int i = tid; i < 64 * 64; i += 256) {
    int k = i >> 6, c = i & 63;
    reinterpret_cast<float*>(&sW2[k >> 1][c])[k & 1] = Wm[i];
  }
  // stage hn = (neigh + h)/(deg+1) for 128 rows, pair-interleaved
  for (int i = tid; i < 128 * 64; i += 256) {
    int r = i >> 6, f = i & 63;
    int grow = base + r;
    float v = 0.0f;
    if (grow < n_nodes) {
      long o = (long)grow * F + f;
      v = (neigh[o] + h[o]) / (deg[grow] + 1.0f);
    }
    reinterpret_cast<float*>(&sA2[r][f >> 1])[f & 1] = v;
  }
  __syncthreads();

  const int wave = tid >> 5;
  const int lane = tid & 31;
  const int m    = lane & 15;   // row (A/D) or col (B/D) index inside tile
  const int hi   = lane >> 4;   // lane-half: K pair {0,1} vs {2,3}; M vs M+8 in D

  v8f acc0 = {}, acc1 = {}, acc2 = {}, acc3 = {};
  const v2f* arow = &sA2[wave * 16 + m][0];

#pragma unroll
  for (int k4 = 0; k4 < 16; ++k4) {
    int k2 = 2 * k4 + hi;       // pair index shared by A and B fragments
    v2f a  = arow[k2];
    v2f b0 = sW2[k2][m];
    v2f b1 = sW2[k2][16 + m];
    v2f b2 = sW2[k2][32 + m];
    v2f b3 = sW2[k2][48 + m];
    acc0 = __builtin_amdgcn_wmma_f32_16x16x4_f32(false, a, false, b0, (short)0, acc0, false, false);
    acc1 = __builtin_amdgcn_wmma_f32_16x16x4_f32(false, a, false, b1, (short)0, acc1, false, false);
    acc2 = __builtin_amdgcn_wmma_f32_16x16x4_f32(false, a, false, b2, (short)0, acc2, false, false);
    acc3 = __builtin_amdgcn_wmma_f32_16x16x4_f32(false, a, false, b3, (short)0, acc3, false, false);
  }

  v8f accs[4] = {acc0, acc1, acc2, acc3};
#pragma unroll
  for (int n = 0; n < 4; ++n) {
#pragma unroll
    for (int r = 0; r < 8; ++r) {
      int row = base + wave * 16 + r + 8 * hi;  // D layout: VGPR r -> M=r (+8 for hi half)
      int col = n * 16 + m;
      if (row < n_nodes) {
        float v = accs[n][r] + bias[col];
        v = fmaxf(v, 0.0f);
        if (!readout) out[(long)row * F + col] = v;
        else          atomicAdd(&hg_sum[gids[row] * F + col], v);
      }
    }
  }
}

// ---------------- classifier head: (hg_sum/count) @ Wc + bc ----------------
__global__ void k_head(const float* __restrict__ hg, const float* __restrict__ counts,
                       const float* __restrict__ Wc, const float* __restrict__ bc,
                       float* __restrict__ outp, int G, int NC) {
  int t = blockIdx.x * blockDim.x + threadIdx.x;
  if (t >= G * NC) return;
  int g = t / NC, c = t % NC;
  float inv = 1.0f / fmaxf(counts[g], 1.0f);
  float acc = bc[c];
  for (int f = 0; f < F; ++f)
    acc += (hg[g * F + f] * inv) * Wc[f * NC + c];
  outp[t] = acc;
}

extern "C" void kernel_launch(void* const* d_in, const int* in_sizes, int n_in,
                              void* d_out, int out_size, void* d_ws, size_t ws_size,
                              hipStream_t stream) {
  const float* in_feat = (const float*)d_in[0];
  const float* ew      = (const float*)d_in[1];
  const float* W1      = (const float*)d_in[2];
  const float* b1      = (const float*)d_in[3];
  const float* W2      = (const float*)d_in[4];
  const float* b2      = (const float*)d_in[5];
  const float* Wc      = (const float*)d_in[6];
  const float* bc      = (const float*)d_in[7];
  const int*   esrc    = (const int*)d_in[8];
  const int*   edst    = (const int*)d_in[9];
  const int*   gids    = (const int*)d_in[10];

  const int N  = in_sizes[0] / F;
  const int E  = in_sizes[1];
  const int NC = in_sizes[6] / F;       // Wc is [64, NC]
  const int G  = out_size / NC;

  // workspace layout (floats)
  float* ws     = (float*)d_ws;
  float* deg_w  = ws;                ws += N;
  float* deg    = ws;                ws += N;
  float* hg_sum = ws;                ws += (long)G * F;
  float* counts = ws;                ws += G;
  float* neigh  = ws;                ws += (long)N * F;
  float* h1     = ws;                /* N*F */

  const int ZB = 2048;
  dim3 blk(256);

  // 1) zero deg_w, deg, hg_sum, counts (contiguous region)
  k_zero<<<ZB, blk, 0, stream>>>(deg_w, (long)2 * N + (long)G * F + G);
  // 2) degrees
  k_edge_deg<<<(E + 255) / 256, blk, 0, stream>>>(ew, edst, deg_w, deg, E);

  // ---- layer 1 ----
  k_zero<<<ZB, blk, 0, stream>>>(neigh, (long)N * F);
  k_spmm<<<(E + 7) / 8, blk, 0, stream>>>(in_feat, ew, deg_w, esrc, edst, neigh, E);
  k_update_gemm<<<(N + 127) / 128, blk, 0, stream>>>(in_feat, neigh, deg, W1, b1,
                                                     h1, nullptr, gids, N, 0);

  // ---- layer 2 (epilogue fused with graph readout) ----
  k_zero<<<ZB, blk, 0, stream>>>(neigh, (long)N * F);
  k_spmm<<<(E + 7) / 8, blk, 0, stream>>>(h1, ew, deg_w, esrc, edst, neigh, E);
  k_counts<<<(N + 255) / 256, blk, 0, stream>>>(gids, counts, N);
  k_update_gemm<<<(N + 127) / 128, blk, 0, stream>>>(h1, neigh, deg, W2, b2,
                                                     nullptr, hg_sum, gids, N, 1);

  // ---- head ----
  k_head<<<(G * NC + 127) / 128, dim3(128), 0, stream>>>(hg_sum, counts, Wc, bc,
                                                         (float*)d_out, G, NC);
}